// ProjectRenderScoreModel_24464133718097
// MI455X (gfx1250) — compile-verified
//
#include <hip/hip_runtime.h>
#include <hip/hip_bf16.h>

// ---------------------------------------------------------------------------
// ProjectRenderScoreModel: gaussian blob splatting.
//   blobs[b,v,n,y,x] = exp(-(dst/(2*sig^2))^e)
//   masks[b,v,y,x]   = min(max_n blobs*inten, 1)
// dst is a rank-4 bilinear form:
//   dst = [p2,-2px,-2py,1] . [1, x, y, x^2+y^2]
// -> one V_WMMA_F32_16X16X4_F32 per 16(points)x16(pixels) tile computes 256
// exact f32 squared distances. Output-stream bound (~203 MB of stores, floor
// ~9us @ 23.3 TB/s); blobs use non-temporal stores since the stream exceeds
// the 192 MB L2. The e==1 fast path is a wave-uniform branch so the common
// case pays 1 transcendental per element instead of 3.
// ---------------------------------------------------------------------------

typedef __attribute__((ext_vector_type(2))) float v2f;
typedef __attribute__((ext_vector_type(8))) float v8f;

#define IMG    256
#define SPIX   (IMG * IMG)          // 65536 pixels
#define BSZ    2
#define NVIEW  3
#define NPTS   128
#define NTILEN (NPTS / 16)          // 8 point tiles
#define NPIXT  (SPIX / 16)          // 4096 pixel tiles (16 px, never crosses a row)
#define GRIDX  64                   // pixel-tile chunks
#define TPW    (NPIXT / GRIDX)      // 64 pixel tiles per wave

#define LOG2E  1.44269504088896340f

__global__ void zero_masks_kernel(float* __restrict__ masks, int n) {
    int i = blockIdx.x * blockDim.x + threadIdx.x;
    if (i < n) masks[i] = 0.0f;
}

// FAST: all exponents in the wave are exactly 1.0 -> blob = exp2(d * w),
//       w[r] = -LOG2E / (2 sig^2)  (single mul + v_exp_f32 per element).
// !FAST: generic (t^e) via exp2(e * log2 t), per-element select for e==1 lanes.
template <bool FAST>
__device__ __forceinline__ void pixel_loop(
    int chunk, int mrow, int sel, int nbase, v2f amat,
    const float* __restrict__ w,        // FAST: -LOG2E*inv2s2 ; else inv2s2
    const float* __restrict__ eexp,
    const float* __restrict__ inten,
    float* __restrict__ blobs_bv, float* __restrict__ masks_bv)
{
    const float step = 2.0f / 255.0f;   // linspace(-1,1,256) spacing
    for (int it = 0; it < TPW; ++it) {
        const int tile = chunk * TPW + it;        // 0..4095
        const int xpix = (tile & 15) * 16 + mrow;
        const int ypix = tile >> 4;
        const float xg = fmaf((float)xpix, step, -1.0f);
        const float yg = fmaf((float)ypix, step, -1.0f);

        // B matrix 4x16 f32: col j = pixel, [1, x_j, y, x_j^2 + y^2]
        v2f bmat;
        bmat.x = sel ? yg                  : 1.0f;   // K2 : K0
        bmat.y = sel ? (xg * xg + yg * yg) : xg;     // K3 : K1

        v8f cacc = {0.f, 0.f, 0.f, 0.f, 0.f, 0.f, 0.f, 0.f};
        // D[m, j] = exact squared distance point m <-> pixel j
        v8f d = __builtin_amdgcn_wmma_f32_16x16x4_f32(
            false, amat, false, bmat, (short)0, cacc, false, false);

        const int pix = tile * 16 + mrow;
        float mmax = 0.0f;
#pragma unroll
        for (int r = 0; r < 8; ++r) {
            float bl;
            if (FAST) {
                bl = __builtin_amdgcn_exp2f(d[r] * w[r]);          // exp(-t)
            } else {
                float t = d[r] * w[r];
                float e = eexp[r];
                float a = (e == 1.0f)
                              ? t
                              : __builtin_amdgcn_exp2f(e * __builtin_amdgcn_logf(t));
                bl = __builtin_amdgcn_exp2f(-a * LOG2E);
            }
            int n = nbase + r + sel * 8;
            // streaming store: 201 MB > 192 MB L2 -> non-temporal
            __builtin_nontemporal_store(bl, blobs_bv + (size_t)n * SPIX + pix);
            mmax = fmaxf(mmax, fminf(bl * inten[r], 1.0f));
        }
        // masks: max over n then clamp<=1 == max over n of clamped (monotone).
        // All values >= 0 -> uint bit order == float order.
        atomicMax((unsigned int*)(masks_bv + pix), __float_as_uint(mmax));
    }
}

__global__ __launch_bounds__(32)
void render_blobs_wmma_kernel(const float* __restrict__ points,       // [BSZ,NVIEW,NPTS,2]
                              const float* __restrict__ sigmas,       // [BSZ,NPTS]
                              const float* __restrict__ exponents,    // [BSZ,NPTS]
                              const float* __restrict__ intensities,  // [BSZ,NPTS]
                              const float* __restrict__ cam_sig,      // [BSZ,NVIEW]
                              const float* __restrict__ cam_exp,      // [BSZ,NVIEW]
                              const float* __restrict__ cam_int,      // [BSZ,NVIEW]
                              float* __restrict__ masks,              // [BSZ*NVIEW*SPIX]
                              float* __restrict__ blobs)              // [BSZ*NVIEW*NPTS*SPIX]
{
    const int lane  = threadIdx.x;          // 0..31, wave32
    const int chunk = blockIdx.x;           // 0..GRIDX-1
    const int ntile = blockIdx.y;           // 0..NTILEN-1
    const int bv    = blockIdx.z;           // 0..BSZ*NVIEW-1
    const int b     = bv / NVIEW;
    const int nbase = ntile * 16;

    const int sel  = lane >> 4;             // 0: lanes 0-15 (K0,K1), 1: lanes 16-31 (K2,K3)
    const int mrow = lane & 15;

    // ---- A matrix 16x4 f32: row m = point (nbase+m), [p2, -2px, -2py, 1]
    const float* pp = points + ((size_t)(bv * NPTS) + (nbase + mrow)) * 2;
    float px = pp[0], py = pp[1];
    px = fminf(fmaxf((px - 128.0f) * (1.0f / 128.0f), -1.0f), 1.0f);
    py = fminf(fmaxf((py - 128.0f) * (1.0f / 128.0f), -1.0f), 1.0f);
    v2f amat;
    amat.x = sel ? (-2.0f * py) : (px * px + py * py);  // K2 : K0
    amat.y = sel ? 1.0f         : (-2.0f * px);         // K3 : K1

    // ---- per-lane params for the 8 D rows this lane holds (n = nbase + r + 8*sel)
    const float cs = cam_sig[bv], ce = cam_exp[bv], ci = cam_int[bv];
    float inv2s2[8], eexp[8], inten[8];
    bool lane_slow = false;
#pragma unroll
    for (int r = 0; r < 8; ++r) {
        int n = nbase + r + sel * 8;
        float sg  = sigmas[b * NPTS + n] * cs;
        inv2s2[r] = 1.0f / (2.0f * sg * sg);
        eexp[r]   = exponents[b * NPTS + n] * ce;
        inten[r]  = intensities[b * NPTS + n] * ci;
        lane_slow |= (eexp[r] != 1.0f);
    }

    float* blobs_bv = blobs + (size_t)bv * NPTS * SPIX;
    float* masks_bv = masks + (size_t)bv * SPIX;

    // Wave-uniform fast-path decision: EXEC stays all-1s inside either loop
    // (required by WMMA), and the common case (all exponents == 1) runs with
    // one v_exp_f32 per element instead of log+exp+exp.
    if (!__any(lane_slow)) {
        float w[8];
#pragma unroll
        for (int r = 0; r < 8; ++r) w[r] = -LOG2E * inv2s2[r];
        pixel_loop<true>(chunk, mrow, sel, nbase, amat, w, eexp, inten,
                         blobs_bv, masks_bv);
    } else {
        pixel_loop<false>(chunk, mrow, sel, nbase, amat, inv2s2, eexp, inten,
                          blobs_bv, masks_bv);
    }
}

extern "C" void kernel_launch(void* const* d_in, const int* in_sizes, int n_in,
                              void* d_out, int out_size, void* d_ws, size_t ws_size,
                              hipStream_t stream) {
    const float* points = (const float*)d_in[0];
    const float* sigmas = (const float*)d_in[1];
    const float* expo   = (const float*)d_in[2];
    const float* inten  = (const float*)d_in[3];
    const float* csig   = (const float*)d_in[4];
    const float* cexp   = (const float*)d_in[5];
    const float* cint   = (const float*)d_in[6];

    float* masks = (float*)d_out;                         // [2*3*65536]
    float* blobs = masks + (size_t)BSZ * NVIEW * SPIX;    // [2*3*128*65536]

    const int nm = BSZ * NVIEW * SPIX;
    zero_masks_kernel<<<(nm + 255) / 256, 256, 0, stream>>>(masks, nm);

    dim3 grid(GRIDX, NTILEN, BSZ * NVIEW);
    render_blobs_wmma_kernel<<<grid, 32, 0, stream>>>(
        points, sigmas, expo, inten, csig, cexp, cint, masks, blobs);
}